// PyTorchChamferLoss_22170621181985
// MI455X (gfx1250) — compile-verified
//
#include <hip/hip_runtime.h>
#include <hip/hip_bf16.h>

typedef float v2f __attribute__((ext_vector_type(2)));
typedef float v8f __attribute__((ext_vector_type(8)));
typedef unsigned int u32x4 __attribute__((ext_vector_type(4)));
typedef int i32x4 __attribute__((ext_vector_type(4)));
typedef int i32x8 __attribute__((ext_vector_type(8)));

#define BB 4
#define PP 4
#define VV 4096
#define MM 2048
#define BIGF 1e10f
#define VROWS 128
#define NCHUNK (VV / VROWS)          /* 32  */
#define NBLOCKS_A (BB * PP * NCHUNK) /* 512 */

#if defined(__has_builtin)
#  if __has_builtin(__builtin_amdgcn_tensor_load_to_lds) && \
      __has_builtin(__builtin_amdgcn_s_wait_tensorcnt)
#    define HAVE_TDM 1
#  endif
#endif
#ifndef HAVE_TDM
#  define HAVE_TDM 0
#endif

// ---------------------------------------------------------------------------
// Kernel 0: init column-min workspace to BIG (uint bit patterns; all values
// written later are >= 0 so unsigned min == float min).
// ---------------------------------------------------------------------------
__global__ void chamfer_init(unsigned* __restrict__ chamY) {
    int i = blockIdx.x * blockDim.x + threadIdx.x;
    if (i < BB * PP * MM) chamY[i] = __float_as_uint(BIGF);
}

// ---------------------------------------------------------------------------
// Kernel 1: main. One wave = one 16-row tile, sweeping valid M columns in
// 16-col tiles. d2 tile = V_WMMA_F32_16X16X4_F32 via the augmented GEMM:
//   A[v] = [px, py, px^2+py^2, 1],  B[c] = [-2ex, -2ey, 1, ex^2+ey^2]
// Edgemaps staged to LDS by the Tensor Data Mover; B operands prebuilt in
// LDS. Inner loop: ds_load_b64 + wmma + med3/min trees + ds_min_u32.
// Clamp + row mask fold into ONE v_med3_num_f32: med3(acc, rb, BIG) with
// rb in {0, BIG}; rmin clamp deferred past the loop (clamp commutes w/ min).
// ---------------------------------------------------------------------------
__global__ __launch_bounds__(256) void chamfer_main(
    const float* __restrict__ vertices,        // (B,V,3)
    const float* __restrict__ projmats,        // (P,3,4)
    const float* __restrict__ edgemaps,        // (B,P,M,2)
    const unsigned char* __restrict__ bmask,   // (B,P,V) bool
    const int* __restrict__ elen,              // (B,P)
    unsigned* __restrict__ chamY,              // (B,P,M) uint(float) mins
    float* __restrict__ sumXpart,              // (NBLOCKS_A)
    float* __restrict__ cntXpart)              // (NBLOCKS_A)
{
    __shared__ v2f      sB[2][MM];   // ready WMMA B-operands (32KB)
    __shared__ unsigned sY[MM];      // per-block column mins (8KB)
#if HAVE_TDM
    __shared__ float    sEM[MM * 2]; // raw edgemaps staged by TDM (16KB)
#endif
    __shared__ float sRed[16];
    __shared__ float sCnt[8];

    const int tid  = threadIdx.x;
    const int wave = tid >> 5;
    const int lane = tid & 31;
    const int half = lane >> 4;
    const int idx  = lane & 15;

    const int bp    = blockIdx.x / NCHUNK;
    const int chunk = blockIdx.x % NCHUNK;
    const int b = bp / PP;
    const int p = bp % PP;

    const float* embase = edgemaps + (size_t)bp * MM * 2;

#if HAVE_TDM
    // ---- Tensor Data Mover: stage 16KB of edgemaps (4096 f32, 1 row) ----
    if (tid == 0) {
        unsigned long long ga = (unsigned long long)(uintptr_t)embase;
        unsigned lds_addr = (unsigned)(uintptr_t)&sEM[0];
        u32x4 g0;
        g0.x = 1u;                                   // count=1, user mode
        g0.y = lds_addr;                             // LDS dest
        g0.z = (unsigned)(ga & 0xffffffffu);         // global_addr[31:0]
        g0.w = (unsigned)((ga >> 32) & 0x01ffffffu)  // global_addr[56:32]
             | (2u << 30);                           // type=2 ("image")
        i32x8 g1;
        g1[0] = (2 << 16);                  // data_size = 4 bytes
        g1[1] = (int)((MM * 2) << 16);      // tensor_dim0[15:0] in [31:16]
        g1[2] = (int)(((MM * 2) >> 16) | (1u << 16)); // dim0 hi | tensor_dim1=1
        g1[3] = (int)((MM * 2) << 16);      // tile_dim0 = 4096 in [31:16]
        g1[4] = 1;                          // tile_dim1 = 1, tile_dim2 = 0
        g1[5] = MM * 2;                     // tensor_dim0_stride low
        g1[6] = 0;
        g1[7] = 0;
        i32x4 gz4 = {0, 0, 0, 0};
        i32x8 gz8 = {0, 0, 0, 0, 0, 0, 0, 0};
        __builtin_amdgcn_tensor_load_to_lds(g0, g1, gz4, gz4, gz8, 0);
    }
#endif

    for (int i = tid; i < MM; i += 256) sY[i] = __float_as_uint(BIGF);

#if HAVE_TDM
    if (tid == 0) __builtin_amdgcn_s_wait_tensorcnt(0);
#endif
    __syncthreads();

    // ---- build WMMA B operands in LDS: sB[0]=K0,K1 rows, sB[1]=K2,K3 ----
    for (int c = tid; c < MM; c += 256) {
#if HAVE_TDM
        const float ex = sEM[2 * c], ey = sEM[2 * c + 1];
#else
        const float ex = embase[2 * c], ey = embase[2 * c + 1];
#endif
        v2f t0; t0.x = -2.0f * ex; t0.y = -2.0f * ey;
        v2f t1; t1.x = 1.0f;       t1.y = ex * ex + ey * ey;
        sB[0][c] = t0;
        sB[1][c] = t1;
    }

    // ---- per-lane A operand (16x4 f32 layout) ----
    const float* pm = projmats + p * 12;
    const float m00=pm[0],m01=pm[1],m02=pm[2], m03=pm[3];
    const float m10=pm[4],m11=pm[5],m12=pm[6], m13=pm[7];
    const float m20=pm[8],m21=pm[9],m22=pm[10],m23=pm[11];

    const int row0 = chunk * VROWS + wave * 16;
    const int v    = row0 + idx;

    const float* vp = vertices + ((size_t)b * VV + v) * 3;
    const float vx = vp[0], vy = vp[1], vz = vp[2];
    const float w  = m20*vx + m21*vy + m22*vz + m23;
    const float px = (m00*vx + m01*vy + m02*vz + m03) / w;
    const float py = (m10*vx + m11*vy + m12*vz + m13) / w;
    const float r2 = px*px + py*py;

    v2f a;
    a.x = half ? r2   : px;   // lanes 0-15: K0 ; lanes 16-31: K2
    a.y = half ? 1.0f : py;   // lanes 0-15: K1 ; lanes 16-31: K3

    const unsigned char bm = bmask[(size_t)bp * VV + v];
    const unsigned rowmask = (unsigned)__ballot(bm != 0) & 0xffffu;

    // rb[j]: 0 if this lane-half's row j is xmask-valid, BIG otherwise.
    // med3(acc, rb, BIG): rb=0   -> clamp(acc, 0, BIG)  (valid row)
    //                     rb=BIG -> BIG                 (masked row)
    float rb[8];
    #pragma unroll
    for (int j = 0; j < 8; ++j)
        rb[j] = ((rowmask >> (j + 8 * half)) & 1u) ? 0.0f : BIGF;

    int len = elen[bp]; if (len > MM) len = MM;
    const int mtFull = len >> 4;          // tiles with all columns valid
    const int mtEnd  = (len + 15) >> 4;   // columns >= len contribute nothing

    float rmin[8];
    #pragma unroll
    for (int j = 0; j < 8; ++j) rmin[j] = BIGF;

    __syncthreads();

    // ---- steady-state loop ----
    for (int mt = 0; mt < mtFull; ++mt) {
        const int c = (mt << 4) + idx;
        const v2f bbv = sB[half][c];
        v8f acc = {};
        acc = __builtin_amdgcn_wmma_f32_16x16x4_f32(
            false, a, false, bbv, (short)0, acc, false, false);

        float q[8];
        #pragma unroll
        for (int j = 0; j < 8; ++j) {
            q[j] = __builtin_amdgcn_fmed3f(acc[j], rb[j], BIGF); // clamp+mask, 1 op
            rmin[j] = fminf(rmin[j], acc[j]);                    // cham_x (clamped later)
        }
        // balanced min tree -> v_min3 friendly, short dependency chain
        const float cmin = fminf(fminf(fminf(q[0], q[1]), fminf(q[2], q[3])),
                                 fminf(fminf(q[4], q[5]), fminf(q[6], q[7])));
        atomicMin(&sY[c], __float_as_uint(cmin)); // both halves; HW merges
    }

    // ---- boundary tile (uniform condition; keeps EXEC all-1 for WMMA) ----
    if (mtFull < mtEnd) {
        const int c = (mtFull << 4) + idx;
        const v2f bbv = sB[half][c];
        v8f acc = {};
        acc = __builtin_amdgcn_wmma_f32_16x16x4_f32(
            false, a, false, bbv, (short)0, acc, false, false);
        const bool cvalid = (c < len);
        float q[8];
        #pragma unroll
        for (int j = 0; j < 8; ++j) {
            q[j] = __builtin_amdgcn_fmed3f(acc[j], rb[j], BIGF);
            if (cvalid) rmin[j] = fminf(rmin[j], acc[j]);
        }
        const float cmin = fminf(fminf(fminf(q[0], q[1]), fminf(q[2], q[3])),
                                 fminf(fminf(q[4], q[5]), fminf(q[6], q[7])));
        atomicMin(&sY[c], __float_as_uint(cmin)); // cols >= len never read back
    }

    // deferred clamp: min_c max(d,0) == max(min_c d, 0)
    #pragma unroll
    for (int j = 0; j < 8; ++j) rmin[j] = fmaxf(rmin[j], 0.0f);

    // ---- cham_x: min across the 16 column classes within each half ----
    #pragma unroll
    for (int m = 1; m <= 8; m <<= 1) {
        #pragma unroll
        for (int j = 0; j < 8; ++j)
            rmin[j] = fminf(rmin[j], __shfl_xor(rmin[j], m, 32));
    }

    if (idx == 0) {   // half0 holds rows 0-7, half1 rows 8-15
        float s = 0.0f;
        #pragma unroll
        for (int j = 0; j < 8; ++j)
            if ((rowmask >> (j + 8 * half)) & 1u) s += rmin[j];
        sRed[wave * 2 + half] = s;
        if (half == 0) sCnt[wave] = (float)__popc(rowmask);
    }
    __syncthreads();

    if (tid == 0) {   // deterministic per-block partials
        float s = 0.0f, cnt = 0.0f;
        #pragma unroll
        for (int i = 0; i < 16; ++i) s += sRed[i];
        #pragma unroll
        for (int i = 0; i < 8; ++i) cnt += sCnt[i];
        sumXpart[blockIdx.x] = s;
        cntXpart[blockIdx.x] = cnt;
    }

    // ---- merge block-local column mins (order-independent) ----
    for (int i = tid; i < MM; i += 256)
        atomicMin(&chamY[(size_t)bp * MM + i], sY[i]);
}

// ---------------------------------------------------------------------------
// Kernel 2: finalize. One block per mesh b; fixed-order reductions only.
// ---------------------------------------------------------------------------
__global__ __launch_bounds__(256) void chamfer_final(
    const unsigned* __restrict__ chamY,
    const float* __restrict__ sumXpart,
    const float* __restrict__ cntXpart,
    const int* __restrict__ elen,
    float* __restrict__ out)
{
    __shared__ float red[256];
    const int tid = threadIdx.x;
    const int b   = blockIdx.x;
    float loss = 0.0f;

    for (int p = 0; p < PP; ++p) {
        const int bp = b * PP + p;
        int len = elen[bp]; if (len > MM) len = MM;

        float s = 0.0f;
        for (int c = tid; c < len; c += 256)
            s += __uint_as_float(chamY[(size_t)bp * MM + c]);
        red[tid] = s;
        __syncthreads();
        for (int off = 128; off > 0; off >>= 1) {
            if (tid < off) red[tid] += red[tid + off];
            __syncthreads();
        }
        if (tid == 0) {
            float sx = 0.0f, cx = 0.0f;
            for (int k = 0; k < NCHUNK; ++k) {
                sx += sumXpart[bp * NCHUNK + k];
                cx += cntXpart[bp * NCHUNK + k];
            }
            const float ylen = (float)(len < 1 ? 1 : len);
            loss += sx / fmaxf(cx, 1.0f) + red[0] / ylen;
        }
        __syncthreads();
    }
    if (tid == 0) out[b] = loss / (float)PP;
}

// ---------------------------------------------------------------------------
extern "C" void kernel_launch(void* const* d_in, const int* in_sizes, int n_in,
                              void* d_out, int out_size, void* d_ws, size_t ws_size,
                              hipStream_t stream) {
    const float*         vertices = (const float*)d_in[0];
    const float*         projmats = (const float*)d_in[1];
    const float*         edgemaps = (const float*)d_in[2];
    const unsigned char* bmask    = (const unsigned char*)d_in[3];
    const int*           elen     = (const int*)d_in[4];
    float* out = (float*)d_out;

    // workspace: chamY (B*P*M u32) | sumXpart (512 f32) | cntXpart (512 f32)
    unsigned* chamY    = (unsigned*)d_ws;
    float*    sumXpart = (float*)(chamY + BB * PP * MM);
    float*    cntXpart = sumXpart + NBLOCKS_A;

    chamfer_init<<<(BB * PP * MM + 255) / 256, 256, 0, stream>>>(chamY);
    chamfer_main<<<NBLOCKS_A, 256, 0, stream>>>(vertices, projmats, edgemaps,
                                                bmask, elen, chamY,
                                                sumXpart, cntXpart);
    chamfer_final<<<BB, 256, 0, stream>>>(chamY, sumXpart, cntXpart, elen, out);
}